// MultiBoxLoss_23502061044409
// MI455X (gfx1250) — compile-verified
//
#include <hip/hip_runtime.h>
#include <stdint.h>

#define P_TOTAL 21824
#define BATCH   128
#define M_OBJ   16
#define OVTH    0.35f
#define TILE    256

// ---------------------------------------------------------------------------
// CDNA5 helpers: async global->LDS DMA (tracked by ASYNCcnt)
// ---------------------------------------------------------------------------
__device__ __forceinline__ uint32_t lds_off32(const void* p) {
  // generic LDS pointer: low 32 bits == wave-relative LDS byte address
  return (uint32_t)(size_t)p;
}

__device__ __forceinline__ void async_copy_b128(uint32_t lds_addr, uint32_t byte_off,
                                                const void* sbase) {
  // GVS mode: mem = SGPR64 base + per-lane 32-bit offset; dest = per-lane LDS addr
  asm volatile("global_load_async_to_lds_b128 %0, %1, %2"
               :: "v"(lds_addr), "v"(byte_off), "s"(sbase) : "memory");
}
__device__ __forceinline__ void wait_async_le1() {
  asm volatile("s_wait_asynccnt 1" ::: "memory");
}
__device__ __forceinline__ void wait_async_le0() {
  asm volatile("s_wait_asynccnt 0" ::: "memory");
}

__device__ __forceinline__ float smooth_l1(float d) {
  float a = fabsf(d);
  return (a < 1.0f) ? 0.5f * a * a : a - 0.5f;
}

// ---------------------------------------------------------------------------
// Kernel 1: per image, argmax over priors for each of the 16 truths
// (tie-break: lowest prior index, matching jnp.argmax first-occurrence)
// ---------------------------------------------------------------------------
__global__ __launch_bounds__(256) void k_best_prior(
    const float* __restrict__ gt, const float* __restrict__ priors,
    int* __restrict__ bpi_out)
{
  __shared__ float t_x0[M_OBJ], t_y0[M_OBJ], t_x1[M_OBJ], t_y1[M_OBJ], t_ar[M_OBJ];
  __shared__ float red_v[M_OBJ][256];
  __shared__ int   red_i[M_OBJ][256];

  const int img = blockIdx.x, tid = threadIdx.x;
  if (tid < M_OBJ) {
    const float* g = gt + ((size_t)img * M_OBJ + tid) * 4;
    float x0 = g[0], y0 = g[1], x1 = g[2], y1 = g[3];
    t_x0[tid] = x0; t_y0[tid] = y0; t_x1[tid] = x1; t_y1[tid] = y1;
    t_ar[tid] = (x1 - x0) * (y1 - y0);
  }
  __syncthreads();

  float bv[M_OBJ]; int bi[M_OBJ];
#pragma unroll
  for (int m = 0; m < M_OBJ; ++m) { bv[m] = -1.0f; bi[m] = 0; }

  for (int p = tid; p < P_TOTAL; p += 256) {
    float4 pr = ((const float4*)priors)[p];
    float px0 = pr.x - 0.5f * pr.z, py0 = pr.y - 0.5f * pr.w;
    float px1 = pr.x + 0.5f * pr.z, py1 = pr.y + 0.5f * pr.w;
    float area_p = pr.z * pr.w;
#pragma unroll
    for (int m = 0; m < M_OBJ; ++m) {
      float iw = fminf(t_x1[m], px1) - fmaxf(t_x0[m], px0);
      float ih = fminf(t_y1[m], py1) - fmaxf(t_y0[m], py0);
      iw = fmaxf(iw, 0.0f); ih = fmaxf(ih, 0.0f);
      float inter = iw * ih;
      float iou = inter / (t_ar[m] + area_p - inter);
      if (iou > bv[m]) { bv[m] = iou; bi[m] = p; }   // keeps smallest p in stripe
    }
  }

#pragma unroll
  for (int m = 0; m < M_OBJ; ++m) { red_v[m][tid] = bv[m]; red_i[m][tid] = bi[m]; }
  __syncthreads();
  for (int off = 128; off > 0; off >>= 1) {
    if (tid < off) {
#pragma unroll
      for (int m = 0; m < M_OBJ; ++m) {
        float a = red_v[m][tid], b = red_v[m][tid + off];
        int ia = red_i[m][tid], ib = red_i[m][tid + off];
        if (b > a || (b == a && ib < ia)) { red_v[m][tid] = b; red_i[m][tid] = ib; }
      }
    }
    __syncthreads();
  }
  if (tid < M_OBJ) bpi_out[img * M_OBJ + tid] = red_i[tid][0];
}

// ---------------------------------------------------------------------------
// Kernel 2: per image, per-prior match (with override), loc loss over
// positives, CE; writes monotone uint32 keys for negatives (0 for positives).
// Priors are staged through LDS with double-buffered async DMA.
// ---------------------------------------------------------------------------
__global__ __launch_bounds__(256) void k_match_ce(
    const float* __restrict__ loc, const float* __restrict__ conf,
    const float* __restrict__ gt, const int* __restrict__ labels,
    const float* __restrict__ priors, const int* __restrict__ bpi,
    uint32_t* __restrict__ keys, float* __restrict__ lossl_out,
    float* __restrict__ cepos_out, int* __restrict__ npos_out)
{
  __shared__ float t_x0[M_OBJ], t_y0[M_OBJ], t_x1[M_OBJ], t_y1[M_OBJ], t_ar[M_OBJ];
  __shared__ int   t_lab[M_OBJ], s_bpi[M_OBJ];
  __shared__ float tile[2][TILE * 4];
  __shared__ float red_f[256];
  __shared__ int   red_n[256];

  const int img = blockIdx.x, tid = threadIdx.x;
  if (tid < M_OBJ) {
    const float* g = gt + ((size_t)img * M_OBJ + tid) * 4;
    float x0 = g[0], y0 = g[1], x1 = g[2], y1 = g[3];
    t_x0[tid] = x0; t_y0[tid] = y0; t_x1[tid] = x1; t_y1[tid] = y1;
    t_ar[tid] = (x1 - x0) * (y1 - y0);
    t_lab[tid] = labels[img * M_OBJ + tid];
    s_bpi[tid] = bpi[img * M_OBJ + tid];
  }

  const int ntiles = (P_TOTAL + TILE - 1) / TILE;   // 86 (last tile: 64 valid)
  // Prologue: async-load tile 0 (clamped index -> uniform issue, no divergence)
  {
    int p = tid; if (p > P_TOTAL - 1) p = P_TOTAL - 1;
    async_copy_b128(lds_off32(&tile[0][tid * 4]), (uint32_t)(p * 16), priors);
  }
  __syncthreads();   // truths visible

  float accL = 0.0f, accC = 0.0f; int accN = 0;

  for (int t = 0; t < ntiles; ++t) {
    const int cur = t & 1, nxt = cur ^ 1;
    if (t + 1 < ntiles) {
      int p = (t + 1) * TILE + tid; if (p > P_TOTAL - 1) p = P_TOTAL - 1;
      async_copy_b128(lds_off32(&tile[nxt][tid * 4]), (uint32_t)(p * 16), priors);
      wait_async_le1();   // in-order: tile t is done, tile t+1 may be in flight
    } else {
      wait_async_le0();
    }
    __syncthreads();

    const int p = t * TILE + tid;
    if (p < P_TOTAL) {
      float pcx = tile[cur][tid * 4 + 0], pcy = tile[cur][tid * 4 + 1];
      float pw  = tile[cur][tid * 4 + 2], ph  = tile[cur][tid * 4 + 3];
      float px0 = pcx - 0.5f * pw, py0 = pcy - 0.5f * ph;
      float px1 = pcx + 0.5f * pw, py1 = pcy + 0.5f * ph;
      float area_p = pw * ph;

      float bv = -1.0f; int bidx = 0;
#pragma unroll
      for (int m = 0; m < M_OBJ; ++m) {   // first-max tie-break (strict >)
        float iw = fminf(t_x1[m], px1) - fmaxf(t_x0[m], px0);
        float ih = fminf(t_y1[m], py1) - fmaxf(t_y0[m], py0);
        iw = fmaxf(iw, 0.0f); ih = fmaxf(ih, 0.0f);
        float inter = iw * ih;
        float iou = inter / (t_ar[m] + area_p - inter);
        if (iou > bv) { bv = iou; bidx = m; }
      }
#pragma unroll
      for (int m = 0; m < M_OBJ; ++m)     // override: last write wins
        if (p == s_bpi[m]) { bidx = m; bv = 2.0f; }

      int lab = (bv < OVTH) ? 0 : t_lab[bidx];

      // prefetch next conf cacheline while we compute
      __builtin_prefetch(conf + ((size_t)img * P_TOTAL + (size_t)p + 256) * 2, 0, 1);

      float c0 = conf[((size_t)img * P_TOTAL + p) * 2 + 0];
      float c1 = conf[((size_t)img * P_TOTAL + p) * 2 + 1];
      float mx = fmaxf(c0, c1), mn = fminf(c0, c1);
      float lse = mx + log1pf(expf(mn - mx));

      uint32_t key;
      if (lab > 0) {
        accC += lse - c1;   // picked = conf[..., 1] (NUM_CLASSES==2, labels==1)
        accN += 1;
        // encode loc target and accumulate smooth-L1
        float mx0 = t_x0[bidx], my0 = t_y0[bidx], mx1v = t_x1[bidx], my1 = t_y1[bidx];
        float gcx = ((mx0 + mx1v) * 0.5f - pcx) / (0.1f * pw);
        float gcy = ((my0 + my1) * 0.5f - pcy) / (0.1f * ph);
        float gw = logf((mx1v - mx0) / pw) * 5.0f;    // /VAR1 (0.2)
        float gh = logf((my1 - my0) / ph) * 5.0f;
        float4 ld = ((const float4*)loc)[(size_t)img * P_TOTAL + p];
        accL += smooth_l1(ld.x - gcx) + smooth_l1(ld.y - gcy) +
                smooth_l1(ld.z - gw)  + smooth_l1(ld.w - gh);
        key = 0u;
      } else {
        float ce = fmaxf(lse - c0, 0.0f);             // >= 0; bits are monotone
        key = __float_as_uint(ce);
      }
      keys[(size_t)img * P_TOTAL + p] = key;
    }
    __syncthreads();   // protect buffer about to be refilled next iteration
  }

  // block reductions
  red_f[tid] = accL; __syncthreads();
  for (int off = 128; off > 0; off >>= 1) { if (tid < off) red_f[tid] += red_f[tid + off]; __syncthreads(); }
  if (tid == 0) lossl_out[img] = red_f[0];
  __syncthreads();
  red_f[tid] = accC; __syncthreads();
  for (int off = 128; off > 0; off >>= 1) { if (tid < off) red_f[tid] += red_f[tid + off]; __syncthreads(); }
  if (tid == 0) cepos_out[img] = red_f[0];
  __syncthreads();
  red_n[tid] = accN; __syncthreads();
  for (int off = 128; off > 0; off >>= 1) { if (tid < off) red_n[tid] += red_n[tid + off]; __syncthreads(); }
  if (tid == 0) npos_out[img] = red_n[0];
}

// ---------------------------------------------------------------------------
// Kernel 3: per image, 4-pass MSB radix-select of the top-j negative CE keys,
// then sum of selected = sum(v > T) + rem * T (exact under ties).
// ---------------------------------------------------------------------------
__global__ __launch_bounds__(256) void k_select(
    const uint32_t* __restrict__ keys, const int* __restrict__ npos,
    const float* __restrict__ cepos, float* __restrict__ lossc_out)
{
  __shared__ uint32_t hist[256];
  __shared__ uint32_t s_prefix, s_rem;
  __shared__ float    s_red[256];

  const int img = blockIdx.x, tid = threadIdx.x;
  const uint32_t* kk = keys + (size_t)img * P_TOTAL;
  const int np = npos[img];
  int j = 3 * np;
  if (j > P_TOTAL - 1) j = P_TOTAL - 1;
  if (j > P_TOTAL - np) j = P_TOTAL - np;   // at most #negatives
  if (j < 0) j = 0;

  float negsum = 0.0f;
  if (j > 0) {
    uint32_t prefix = 0, rem = (uint32_t)j;
    for (int shift = 24; shift >= 0; shift -= 8) {
      const uint32_t himask = (shift == 24) ? 0u : (0xFFFFFFFFu << (shift + 8));
      hist[tid] = 0;
      __syncthreads();
      for (int p = tid; p < P_TOTAL; p += 256) {
        uint32_t v = kk[p];
        if ((v & himask) == (prefix & himask))
          atomicAdd(&hist[(v >> shift) & 255u], 1u);
      }
      __syncthreads();
      if (tid == 0) {
        uint32_t cum = 0, newrem = rem; int sel = 0;
        for (int b = 255; b >= 0; --b) {
          uint32_t c = hist[b];
          if (cum + c >= rem) { sel = b; newrem = rem - cum; break; }
          cum += c;
        }
        s_prefix = (prefix & himask) | ((uint32_t)sel << shift);
        s_rem = newrem;
      }
      __syncthreads();
      prefix = s_prefix; rem = s_rem;
      __syncthreads();
    }
    const uint32_t T = prefix;   // exact 32-bit threshold key
    float local = 0.0f;
    for (int p = tid; p < P_TOTAL; p += 256) {
      uint32_t v = kk[p];
      if (v > T) local += __uint_as_float(v);
    }
    s_red[tid] = local; __syncthreads();
    for (int off = 128; off > 0; off >>= 1) { if (tid < off) s_red[tid] += s_red[tid + off]; __syncthreads(); }
    negsum = s_red[0] + (float)rem * __uint_as_float(T);
  }
  if (tid == 0) lossc_out[img] = cepos[img] + negsum;
}

// ---------------------------------------------------------------------------
// Kernel 4: final reduction across the 128 images
// ---------------------------------------------------------------------------
__global__ __launch_bounds__(128) void k_final(
    const float* __restrict__ lossl, const float* __restrict__ lossc,
    const int* __restrict__ npos, float* __restrict__ out)
{
  __shared__ float sl[128], sc[128];
  __shared__ int   sn[128];
  const int t = threadIdx.x;
  sl[t] = lossl[t]; sc[t] = lossc[t]; sn[t] = npos[t];
  __syncthreads();
  for (int off = 64; off > 0; off >>= 1) {
    if (t < off) { sl[t] += sl[t + off]; sc[t] += sc[t + off]; sn[t] += sn[t + off]; }
    __syncthreads();
  }
  if (t == 0) {
    float N = fmaxf((float)sn[0], 1.0f);
    out[0] = sl[0] / N;
    out[1] = sc[0] / N;
  }
}

// ---------------------------------------------------------------------------
extern "C" void kernel_launch(void* const* d_in, const int* in_sizes, int n_in,
                              void* d_out, int out_size, void* d_ws, size_t ws_size,
                              hipStream_t stream) {
  (void)in_sizes; (void)n_in; (void)out_size; (void)ws_size;
  const float* loc    = (const float*)d_in[0];   // (B,P,4) f32
  const float* conf   = (const float*)d_in[1];   // (B,P,2) f32
  const float* gt     = (const float*)d_in[2];   // (B,16,4) f32 point-form
  const int*   labels = (const int*)d_in[3];     // (B,16) i32
  const float* priors = (const float*)d_in[4];   // (P,4) f32 cx,cy,w,h

  uint8_t* ws = (uint8_t*)d_ws;
  uint32_t* keys = (uint32_t*)ws;
  size_t off = (size_t)BATCH * P_TOTAL * sizeof(uint32_t);   // ~11.17 MB
  int*   bpi   = (int*)(ws + off);   off += (size_t)BATCH * M_OBJ * sizeof(int);
  float* lossl = (float*)(ws + off); off += (size_t)BATCH * sizeof(float);
  float* cepos = (float*)(ws + off); off += (size_t)BATCH * sizeof(float);
  float* lossc = (float*)(ws + off); off += (size_t)BATCH * sizeof(float);
  int*   npos  = (int*)(ws + off);   off += (size_t)BATCH * sizeof(int);

  k_best_prior<<<BATCH, 256, 0, stream>>>(gt, priors, bpi);
  k_match_ce  <<<BATCH, 256, 0, stream>>>(loc, conf, gt, labels, priors, bpi,
                                          keys, lossl, cepos, npos);
  k_select    <<<BATCH, 256, 0, stream>>>(keys, npos, cepos, lossc);
  k_final     <<<1, 128, 0, stream>>>(lossl, lossc, npos, (float*)d_out);
}